// OneHotWeightedAverage_63694365000152
// MI455X (gfx1250) — compile-verified
//
#include <hip/hip_runtime.h>

#define B_DIM      256
#define S_DIM      2048
#define V_DIM      128000
#define NTHREADS   256
#define PER_THREAD (S_DIM / NTHREADS)          // 8 elements / thread
#define NWAVES     (NTHREADS / 32)             // 8 wave32s / block
#define ZERO_ITERS (V_DIM / 4 / NTHREADS)      // 125 float4 stores / thread

__device__ __forceinline__ float wave_reduce_max(float v) {
#pragma unroll
  for (int off = 16; off > 0; off >>= 1)
    v = fmaxf(v, __shfl_xor(v, off, 32));
  return v;
}

__device__ __forceinline__ float wave_reduce_sum(float v) {
#pragma unroll
  for (int off = 16; off > 0; off >>= 1)
    v += __shfl_xor(v, off, 32);
  return v;
}

__global__ __launch_bounds__(NTHREADS)
void onehot_softmax_scatter(const float* __restrict__ w_es,
                            const int*   __restrict__ x,
                            float*       __restrict__ out) {
  __shared__ float s_wes[S_DIM];     // staged w_es row (8 KB)
  __shared__ float s_redm[NWAVES];
  __shared__ float s_reds[NWAVES];

  const int b    = blockIdx.x;
  const int tid  = threadIdx.x;
  const int lane = tid & 31;
  const int wid  = tid >> 5;

  // ---- 1) async-stage w_es[b, :] into LDS (32 B per thread, 2x b128) ----
  {
    const float* srow = w_es + (size_t)b * S_DIM;            // uniform -> SGPR pair
    unsigned voff   = (unsigned)(tid * PER_THREAD * sizeof(float));
    unsigned ldsoff = (unsigned)(size_t)&s_wes[tid * PER_THREAD]; // low 32b = LDS offset
    asm volatile(
        "global_load_async_to_lds_b128 %0, %1, %2\n\t"
        "global_load_async_to_lds_b128 %0, %1, %2 offset:16"
        :
        : "v"(ldsoff), "v"(voff), "s"(srow)
        : "memory");
  }

  // ---- 2) zero-fill w_a[b, :] while the async copy is in flight ----
  {
    float4* zrow = (float4*)(out + (size_t)b * V_DIM);
    const float4 z4 = make_float4(0.f, 0.f, 0.f, 0.f);
#pragma unroll 5
    for (int k = 0; k < ZERO_ITERS; ++k)
      zrow[tid + k * NTHREADS] = z4;
  }

  // ---- 3) order zeros in L2, land the LDS row, sync the block ----
  asm volatile("s_wait_storecnt 0" ::: "memory");
  asm volatile("s_wait_asynccnt 0" ::: "memory");
  __syncthreads();

  // ---- 4) softmax over the row ----
  float v[PER_THREAD];
  {
    float4 a = *(const float4*)&s_wes[tid * PER_THREAD];
    float4 c = *(const float4*)&s_wes[tid * PER_THREAD + 4];
    v[0] = a.x; v[1] = a.y; v[2] = a.z; v[3] = a.w;
    v[4] = c.x; v[5] = c.y; v[6] = c.z; v[7] = c.w;
  }

  float m = v[0];
#pragma unroll
  for (int j = 1; j < PER_THREAD; ++j) m = fmaxf(m, v[j]);
  m = wave_reduce_max(m);
  if (lane == 0) s_redm[wid] = m;
  __syncthreads();
  float rowmax = s_redm[0];
#pragma unroll
  for (int i = 1; i < NWAVES; ++i) rowmax = fmaxf(rowmax, s_redm[i]);

  float e[PER_THREAD];
  float s = 0.f;
#pragma unroll
  for (int j = 0; j < PER_THREAD; ++j) { e[j] = __expf(v[j] - rowmax); s += e[j]; }
  s = wave_reduce_sum(s);
  if (lane == 0) s_reds[wid] = s;
  __syncthreads();
  float rowsum = 0.f;
#pragma unroll
  for (int i = 0; i < NWAVES; ++i) rowsum += s_reds[i];
  const float inv = 1.0f / rowsum;

  float w[PER_THREAD];
#pragma unroll
  for (int j = 0; j < PER_THREAD; ++j) w[j] = e[j] * inv;

  // ---- 5a) write weights output (second tuple element) ----
  {
    float* wout = out + (size_t)B_DIM * V_DIM + (size_t)b * S_DIM + tid * PER_THREAD;
    *(float4*)(wout)     = make_float4(w[0], w[1], w[2], w[3]);
    *(float4*)(wout + 4) = make_float4(w[4], w[5], w[6], w[7]);
  }

  // ---- 5b) scatter into w_a[b, :] (hits zeroed lines in 192MB L2) ----
  {
    const int* xrow = x + (size_t)b * S_DIM + tid * PER_THREAD;
    float* arow = out + (size_t)b * V_DIM;
#pragma unroll
    for (int j = 0; j < PER_THREAD; ++j) {
      int idx = xrow[j];
      arow[idx] = w[j];
    }
  }
}

extern "C" void kernel_launch(void* const* d_in, const int* in_sizes, int n_in,
                              void* d_out, int out_size, void* d_ws, size_t ws_size,
                              hipStream_t stream) {
  const float* w_es = (const float*)d_in[0];
  const int*   x    = (const int*)d_in[1];
  float*       out  = (float*)d_out;

  hipLaunchKernelGGL(onehot_softmax_scatter,
                     dim3(B_DIM), dim3(NTHREADS), 0, stream,
                     w_es, x, out);
}